// ChamferDistance_3032246911459
// MI455X (gfx1250) — compile-verified
//
#include <hip/hip_runtime.h>

typedef float v2f __attribute__((ext_vector_type(2)));
typedef float v8f __attribute__((ext_vector_type(8)));

#define CHAMFER_CHUNK 1024  // target points staged in LDS per block iteration (16 KB)

// Build padded point table: (x, y, z, ||p||^2) per point.
__global__ __launch_bounds__(256) void chamfer_pack_kernel(
    const float* __restrict__ src, float4* __restrict__ dst, int total) {
  int i = blockIdx.x * blockDim.x + threadIdx.x;
  if (i < total) {
    float x = src[3 * i + 0];
    float y = src[3 * i + 1];
    float z = src[3 * i + 2];
    dst[i] = make_float4(x, y, z, x * x + y * y + z * z);
  }
}

// For each P row, min over all Q cols of squared distance, via
// V_WMMA_F32_16X16X4_F32 with augmented operands:
//   A[m][:] = [-2px, -2py, -2pz, 1]   (16x4)
//   B[:][n] = [qx, qy, qz, ||q||^2]   (4x16)
//   C[m][n] = ||p_m||^2
// => D[m][n] = ||p-q||^2 exactly.
// One wave owns a 16-row P tile; 8 waves per block share the LDS Q stage.
__global__ __launch_bounds__(256) void chamfer_min_kernel(
    const float4* __restrict__ P4, const float4* __restrict__ Q4,
    float* __restrict__ outmin, int N, int M) {
  __shared__ float4 qs[CHAMFER_CHUNK];

  const int b    = blockIdx.y;
  const int lane = threadIdx.x & 31;
  const int wave = threadIdx.x >> 5;
  const int half = lane >> 4;   // 0: lanes 0-15, 1: lanes 16-31
  const int sub  = lane & 15;
  const int rowbase = blockIdx.x * 128 + wave * 16;

  const float4* __restrict__ Pb = P4 + (size_t)b * N;
  const float4* __restrict__ Qb = Q4 + (size_t)b * M;

  // A operand (16x4 f32 layout): lanes 0-15 hold K=0,1; lanes 16-31 hold K=2,3.
  float4 p = Pb[rowbase + sub];
  v2f amat;
  amat.x = half ? (-2.0f * p.z) : (-2.0f * p.x);  // K0 / K2
  amat.y = half ? 1.0f          : (-2.0f * p.y);  // K1 / K3

  // C init (16x16 f32 layout): VGPR r, lane-half h -> row (r + 8*h).
  v8f cinit;
#pragma unroll
  for (int r = 0; r < 8; ++r)
    cinit[r] = Pb[rowbase + r + 8 * half].w;

  v8f runmin;
#pragma unroll
  for (int r = 0; r < 8; ++r) runmin[r] = 3.4e38f;

  const float* qsf = reinterpret_cast<const float*>(qs);

  for (int c0 = 0; c0 < M; c0 += CHAMFER_CHUNK) {
    __syncthreads();
#pragma unroll
    for (int i = 0; i < CHAMFER_CHUNK / 256; ++i)
      qs[i * 256 + threadIdx.x] = Qb[c0 + i * 256 + threadIdx.x];
    __syncthreads();

#pragma unroll 4
    for (int t = 0; t < CHAMFER_CHUNK; t += 16) {
      // B operand (4x16): lane sub = column; half selects (qx,qy) vs (qz,||q||^2)
      // via an 8-byte offset -> single ds_load_b64, no select ops.
      float2 q2 = *reinterpret_cast<const float2*>(qsf + (t + sub) * 4 + half * 2);
      v2f bmat;
      bmat.x = q2.x;
      bmat.y = q2.y;
      v8f d = __builtin_amdgcn_wmma_f32_16x16x4_f32(
          /*neg_a=*/false, amat, /*neg_b=*/false, bmat,
          /*c_mod=*/(short)0, cinit, /*reuse_a=*/false, /*reuse_b=*/false);
#pragma unroll
      for (int r = 0; r < 8; ++r) runmin[r] = fminf(runmin[r], d[r]);
    }
  }

  // Row-min: reduce across the 16 lanes of each half (wave32 shuffles).
#pragma unroll
  for (int off = 8; off >= 1; off >>= 1) {
#pragma unroll
    for (int r = 0; r < 8; ++r)
      runmin[r] = fminf(runmin[r], __shfl_xor(runmin[r], off, 16));
  }

  if (sub == 0) {
#pragma unroll
    for (int r = 0; r < 8; ++r)
      outmin[(size_t)b * N + rowbase + 8 * half + r] = runmin[r];
  }
}

__global__ __launch_bounds__(256) void chamfer_reduce_kernel(
    const float* __restrict__ dpt, const float* __restrict__ dtp,
    int npt, int ntp, float* __restrict__ out) {
  __shared__ float s[256];
  const float wpt = 1.0f / (float)npt;
  const float wtp = 1.0f / (float)ntp;
  float acc = 0.0f;
  for (int i = threadIdx.x; i < npt; i += 256) acc += dpt[i] * wpt;
  for (int i = threadIdx.x; i < ntp; i += 256) acc += dtp[i] * wtp;
  s[threadIdx.x] = acc;
  __syncthreads();
  for (int o = 128; o > 0; o >>= 1) {
    if (threadIdx.x < o) s[threadIdx.x] += s[threadIdx.x + o];
    __syncthreads();
  }
  if (threadIdx.x == 0) out[0] = s[0];
}

extern "C" void kernel_launch(void* const* d_in, const int* in_sizes, int n_in,
                              void* d_out, int out_size, void* d_ws, size_t ws_size,
                              hipStream_t stream) {
  const float* pred   = (const float*)d_in[0];
  const float* target = (const float*)d_in[1];
  const int B = 4;
  const int N = in_sizes[0] / (B * 3);
  const int M = in_sizes[1] / (B * 3);

  // Workspace carve: P4 | Q4 | d_pt | d_tp  (~1.25 MB total)
  float4* P4 = (float4*)d_ws;
  float4* Q4 = P4 + (size_t)B * N;
  float*  dpt = (float*)(Q4 + (size_t)B * M);
  float*  dtp = dpt + (size_t)B * N;

  const int totalP = B * N;
  const int totalQ = B * M;
  chamfer_pack_kernel<<<(totalP + 255) / 256, 256, 0, stream>>>(pred, P4, totalP);
  chamfer_pack_kernel<<<(totalQ + 255) / 256, 256, 0, stream>>>(target, Q4, totalQ);

  dim3 block(256);
  dim3 gridPT(N / 128, B);
  dim3 gridTP(M / 128, B);
  chamfer_min_kernel<<<gridPT, block, 0, stream>>>(P4, Q4, dpt, N, M);
  chamfer_min_kernel<<<gridTP, block, 0, stream>>>(Q4, P4, dtp, M, N);

  chamfer_reduce_kernel<<<1, 256, 0, stream>>>(dpt, dtp, totalP, totalQ,
                                               (float*)d_out);
}